// DWT_2611340116590
// MI455X (gfx1250) — compile-verified
//
#include <hip/hip_runtime.h>

// 2D Haar DWT, single level, NHWC (8,256,256,64) f32 -> same-shape quadrant layout.
// Pure streaming op: 268 MB traffic, ~11.5us roofline @ 23.3 TB/s. Strategy:
// one thread = one 2x2 spatial block x 4 channels, b128 NT loads/stores,
// wave32-coalesced (16 lanes x float4 = one 256B channel vector).

#define Bn 8
#define Hn 256
#define Wn 256
#define Cn 64

// native clang vector so nontemporal builtins accept it (HIP float4 is a class)
typedef float v4f __attribute__((ext_vector_type(4)));

__global__ __launch_bounds__(256) void haar_dwt2d_nt(const float* __restrict__ x,
                                                     float* __restrict__ out) {
    constexpr int WS = Cn;            // stride between w positions (floats)
    constexpr int HS = Wn * Cn;       // stride between h positions
    constexpr int BS = Hn * Wn * Cn;  // stride between batches
    constexpr int Hh = Hn / 2;
    constexpr int Wh = Wn / 2;

    const int tid = threadIdx.x;
    const int c   = (tid & 15) << 2;              // channel offset, float4 granularity
    const int j   = blockIdx.x * 16 + (tid >> 4); // output column 0..127
    const int i   = blockIdx.y;                   // output row    0..127
    const int b   = blockIdx.z;

    // input 2x2 block base: (b, 2i, 2j, c)
    const int in00 = b * BS + (2 * i) * HS + (2 * j) * WS + c;

    const v4f x00 = __builtin_nontemporal_load(reinterpret_cast<const v4f*>(x + in00));
    const v4f x01 = __builtin_nontemporal_load(reinterpret_cast<const v4f*>(x + in00 + WS));
    const v4f x10 = __builtin_nontemporal_load(reinterpret_cast<const v4f*>(x + in00 + HS));
    const v4f x11 = __builtin_nontemporal_load(reinterpret_cast<const v4f*>(x + in00 + HS + WS));

    // Haar butterfly (vector ops lower to packed f32 VALU)
    const v4f s0 = x00 + x01;  // row 2i   sum
    const v4f d0 = x00 - x01;  // row 2i   diff
    const v4f s1 = x10 + x11;  // row 2i+1 sum
    const v4f d1 = x10 - x11;  // row 2i+1 diff

    const v4f ll = 0.5f * (s0 + s1);
    const v4f lh = 0.5f * (s0 - s1);
    const v4f hl = 0.5f * (d0 + d1);
    const v4f hh = 0.5f * (d0 - d1);

    // quadrant layout: LL (i,j)     | HL (i, j+128)
    //                  LH (i+128,j) | HH (i+128, j+128)
    const int oll = b * BS + i * HS + j * WS + c;
    __builtin_nontemporal_store(ll, reinterpret_cast<v4f*>(out + oll));
    __builtin_nontemporal_store(lh, reinterpret_cast<v4f*>(out + oll + Hh * HS));
    __builtin_nontemporal_store(hl, reinterpret_cast<v4f*>(out + oll + Wh * WS));
    __builtin_nontemporal_store(hh, reinterpret_cast<v4f*>(out + oll + Hh * HS + Wh * WS));
}

extern "C" void kernel_launch(void* const* d_in, const int* in_sizes, int n_in,
                              void* d_out, int out_size, void* d_ws, size_t ws_size,
                              hipStream_t stream) {
    const float* x = reinterpret_cast<const float*>(d_in[0]);
    float* out     = reinterpret_cast<float*>(d_out);

    // grid: (j-tiles of 16, i rows, batch); block: 256 threads = 8 waves (wave32)
    dim3 grid(Wn / 2 / 16, Hn / 2, Bn);
    dim3 block(256);
    haar_dwt2d_nt<<<grid, block, 0, stream>>>(x, out);
}